// FlockingLGTCN_49684181680925
// MI455X (gfx1250) — compile-verified
//
#include <hip/hip_runtime.h>
#include <math.h>

// Problem constants (match reference)
static const int CB = 4, CT = 32, CN = 512;
static const int CIN = 10, CF = 50, CFC = 128, COUT = 2;

typedef __attribute__((ext_vector_type(2))) float v2f;
typedef __attribute__((ext_vector_type(8))) float v8f;

// ---------------------------------------------------------------------------
// Generic batched fp32 GEMM using V_WMMA_F32_16X16X4_F32.
// C[z] = act(A[z] (MxK, lda) @ B[z] (KxN, ldb) + bias[N])
// One wave per 16x32 C strip (two 16x16 WMMA tiles sharing the A fragment).
// Requirements (hold at all call sites): M % 16 == 0, lda even (so the v2f
// A-fragment load is 8B aligned).
// A-frag layout (ISA 7.12.2, 32-bit A 16x4): lanes 0-15 M=lane {K=0,1},
// lanes 16-31 M=lane-16 {K=2,3}. C/D: VGPR r -> row tm + r + 8*(lane>=16),
// col = tn + (lane&15).
// Out-of-range columns are CLAMPED for loads (values land only in accumulator
// columns that are never stored), keeping the hot loop branch-free.
// ---------------------------------------------------------------------------
__global__ __launch_bounds__(128)
void gemm_wmma_f32(const float* __restrict__ A, long long sA, int lda,
                   const float* __restrict__ Bm, long long sB, int ldb,
                   float* __restrict__ C, long long sC, int ldc,
                   const float* __restrict__ bias,
                   int M, int N, int K, int relu) {
  const int lane  = threadIdx.x & 31;
  const int wave  = threadIdx.x >> 5;
  const int npair = (N + 31) >> 5;                 // 32 output cols per wave
  const int tiles = (M >> 4) * npair;
  const int tile  = blockIdx.x * (blockDim.x >> 5) + wave;
  if (tile >= tiles) return;                       // wave-uniform exit
  const int tm = (tile / npair) << 4;
  const int tn = (tile % npair) << 5;

  const float* Ab = A  + (long long)blockIdx.z * sA;
  const float* Bb = Bm + (long long)blockIdx.z * sB;
  float*       Cb = C  + (long long)blockIdx.z * sC;

  const int mrow  = lane & 15;
  const int khalf = (lane >> 4) << 1;              // 0 or 2
  const int n0 = tn + (lane & 15);
  const int n1 = n0 + 16;
  const float* ap  = Ab + (long long)(tm + mrow) * lda + khalf;
  const float* bp0 = Bb + (n0 < N ? n0 : N - 1);   // clamped (see above)
  const float* bp1 = Bb + (n1 < N ? n1 : N - 1);

  v8f acc0 = {}, acc1 = {};
  const int K4 = K & ~3;
  int k0 = 0;
#pragma unroll 2
  for (; k0 < K4; k0 += 4) {
    const v2f av = *(const v2f*)(ap + k0);         // global_load_b64
    const long long kb = (long long)(k0 + khalf) * ldb;
    v2f bv0, bv1;
    bv0.x = bp0[kb];       bv0.y = bp0[kb + ldb];
    bv1.x = bp1[kb];       bv1.y = bp1[kb + ldb];
    acc0 = __builtin_amdgcn_wmma_f32_16x16x4_f32(false, av, false, bv0,
                                                 (short)0, acc0, false, false);
    acc1 = __builtin_amdgcn_wmma_f32_16x16x4_f32(false, av, false, bv1,
                                                 (short)0, acc1, false, false);
  }
  if (k0 < K) {                                    // K tail (K%4==2 here)
    const int ka = k0 + khalf;
    v2f av = {}, bv0 = {}, bv1 = {};
    if (ka < K) {
      av.x  = ap[k0];
      bv0.x = bp0[(long long)ka * ldb];
      bv1.x = bp1[(long long)ka * ldb];
    }
    if (ka + 1 < K) {
      av.y  = ap[k0 + 1];
      bv0.y = bp0[(long long)(ka + 1) * ldb];
      bv1.y = bp1[(long long)(ka + 1) * ldb];
    }
    acc0 = __builtin_amdgcn_wmma_f32_16x16x4_f32(false, av, false, bv0,
                                                 (short)0, acc0, false, false);
    acc1 = __builtin_amdgcn_wmma_f32_16x16x4_f32(false, av, false, bv1,
                                                 (short)0, acc1, false, false);
  }

  const int rbase = tm + ((lane >> 4) << 3);
  if (n0 < N) {
    const float badd = bias ? bias[n0] : 0.0f;
#pragma unroll
    for (int r = 0; r < 8; ++r) {
      float v = acc0[r] + badd;
      if (relu) v = fmaxf(v, 0.0f);
      Cb[(long long)(rbase + r) * ldc + n0] = v;
    }
  }
  if (n1 < N) {
    const float badd = bias ? bias[n1] : 0.0f;
#pragma unroll
    for (int r = 0; r < 8; ++r) {
      float v = acc1[r] + badd;
      if (relu) v = fmaxf(v, 0.0f);
      Cb[(long long)(rbase + r) * ldc + n1] = v;
    }
  }
}

// ---------------------------------------------------------------------------
// dinv[row] = deg>0 ? rsqrt(deg) : 0, deg = sum of one adjacency row.
// One wave per row, coalesced streaming of A (this is the extra full read).
// ---------------------------------------------------------------------------
__global__ void deg_dinv_kernel(const float* __restrict__ A, float* __restrict__ dinv,
                                int rows, int n) {
  const int wid  = (blockIdx.x * blockDim.x + threadIdx.x) >> 5;
  const int lane = threadIdx.x & 31;
  if (wid >= rows) return;
  const float* row = A + (long long)wid * n;
  float s = 0.0f;
  for (int i = lane; i < n; i += 32) s += row[i];
#pragma unroll
  for (int off = 16; off > 0; off >>= 1) s += __shfl_xor(s, off, 32);
  if (lane == 0) dinv[wid] = (s > 0.0f) ? rsqrtf(s) : 0.0f;
}

// Repack Wch(3,2,50,50) + Wcu(3,2,50,50) into Wz(200,150):
// rows [0:50)=h->Wch[c,0], [50:100)=Lh->Wch[c,1], [100:150)=x->Wcu[c,0],
// [150:200)=Lx->Wcu[c,1]; column = c*50+g. (bias = bc flat, used directly)
__global__ void pack_wz_kernel(const float* __restrict__ Wch,
                               const float* __restrict__ Wcu,
                               float* __restrict__ Wz) {
  int idx = blockIdx.x * blockDim.x + threadIdx.x;
  if (idx >= 200 * 150) return;
  int r = idx / 150, cg = idx % 150;
  int c = cg / 50, g = cg % 50;
  int src = r / 50, f = r % 50;
  float v;
  if (src == 0)      v = Wch[((c * 2 + 0) * 50 + f) * 50 + g];
  else if (src == 1) v = Wch[((c * 2 + 1) * 50 + f) * 50 + g];
  else if (src == 2) v = Wcu[((c * 2 + 0) * 50 + f) * 50 + g];
  else               v = Wcu[((c * 2 + 1) * 50 + f) * 50 + g];
  Wz[idx] = v;
}

// Vs[b,n,f] = dinv[b,t,n] * (f<50 ? h[b,n,f] : x[b,n,f-50])
__global__ void prescale_kernel(const float* __restrict__ h, const float* __restrict__ x,
                                const float* __restrict__ dinv, float* __restrict__ Vs,
                                int t) {
  int idx = blockIdx.x * blockDim.x + threadIdx.x;
  if (idx >= CB * CN * 2 * CF) return;
  int row = idx / (2 * CF), f = idx % (2 * CF);
  int b = row / CN, n = row % CN;
  float dv = dinv[((long long)b * CT + t) * CN + n];
  float v = (f < CF) ? h[row * CF + f] : x[row * CF + (f - CF)];
  Vs[idx] = dv * v;
}

// U = [h, h - dinv*W[:, :50], x, x - dinv*W[:, 50:100]]  (B*N, 200)
__global__ void combine_kernel(const float* __restrict__ h, const float* __restrict__ x,
                               const float* __restrict__ W, const float* __restrict__ dinv,
                               float* __restrict__ U, int t) {
  int idx = blockIdx.x * blockDim.x + threadIdx.x;
  if (idx >= CB * CN * 4 * CF) return;
  int row = idx / (4 * CF), f = idx % (4 * CF);
  int b = row / CN, n = row % CN;
  float dv = dinv[((long long)b * CT + t) * CN + n];
  float v;
  if (f < CF)            v = h[row * CF + f];
  else if (f < 2 * CF)   v = h[row * CF + (f - CF)]     - dv * W[row * 2 * CF + (f - CF)];
  else if (f < 3 * CF)   v = x[row * CF + (f - 2 * CF)];
  else                   v = x[row * CF + (f - 3 * CF)] - dv * W[row * 2 * CF + CF + (f - 3 * CF)];
  U[idx] = v;
}

// h_new = sigmoid(z0)*tanh(z1) + (1-sigmoid(z0))*tanh(z2), z row-major ld=150
__global__ void gate_kernel(const float* __restrict__ z, float* __restrict__ h) {
  int idx = blockIdx.x * blockDim.x + threadIdx.x;
  if (idx >= CB * CN * CF) return;
  int row = idx / CF, f = idx % CF;
  float z0 = z[row * 150 + f];
  float z1 = z[row * 150 + CF + f];
  float z2 = z[row * 150 + 2 * CF + f];
  float gt = 1.0f / (1.0f + expf(-z0));
  h[idx] = gt * tanhf(z1) + (1.0f - gt) * tanhf(z2);
}

// ---------------------------------------------------------------------------
static inline void launch_gemm(const float* A, long long sA, int lda,
                               const float* Bm, long long sB, int ldb,
                               float* C, long long sC, int ldc,
                               const float* bias, int M, int N, int K,
                               int batch, int relu, hipStream_t stream) {
  int npair = (N + 31) >> 5;
  int tiles = (M >> 4) * npair;
  dim3 grid((tiles + 3) / 4, 1, batch);
  gemm_wmma_f32<<<grid, 128, 0, stream>>>(A, sA, lda, Bm, sB, ldb,
                                          C, sC, ldc, bias, M, N, K, relu);
}

extern "C" void kernel_launch(void* const* d_in, const int* in_sizes, int n_in,
                              void* d_out, int out_size, void* d_ws, size_t ws_size,
                              hipStream_t stream) {
  const float* obs  = (const float*)d_in[0];   // (B,T,N,IN)
  const float* adj  = (const float*)d_in[1];   // (B,T,N,N)
  const float* h0   = (const float*)d_in[2];   // (B,1,N,F)
  const float* We1  = (const float*)d_in[3];   // (10,128)
  const float* be1  = (const float*)d_in[4];
  const float* We2  = (const float*)d_in[5];   // (128,50)
  const float* be2  = (const float*)d_in[6];
  const float* Wch  = (const float*)d_in[7];   // (3,2,50,50)
  const float* Wcu  = (const float*)d_in[8];
  const float* bc   = (const float*)d_in[9];   // (3,50) -> flat 150 bias
  const float* Wd1  = (const float*)d_in[10];  // (50,128)
  const float* bd1  = (const float*)d_in[11];
  const float* Wd2  = (const float*)d_in[12];  // (128,128)
  const float* bd2  = (const float*)d_in[13];
  const float* Wd3  = (const float*)d_in[14];  // (128,2)
  const float* bd3  = (const float*)d_in[15];
  float* out = (float*)d_out;

  // Workspace carve-up (floats)
  float* ws = (float*)d_ws;
  size_t off = 0;
  float* dinv = ws + off; off += (size_t)CB * CT * CN;          // 65536
  float* h    = ws + off; off += (size_t)CB * CN * CF;          // 102400
  float* x    = ws + off; off += (size_t)CB * CN * CF;          // 102400
  float* tmp1 = ws + off; off += (size_t)CB * CN * CFC;         // 262144
  float* Vs   = ws + off; off += (size_t)CB * CN * 2 * CF;      // 204800
  float* Wb   = ws + off; off += (size_t)CB * CN * 2 * CF;      // 204800
  float* U    = ws + off; off += (size_t)CB * CN * 4 * CF;      // 409600
  float* z    = ws + off; off += (size_t)CB * CN * 150;         // 307200
  float* y1   = ws + off; off += (size_t)CB * CN * CFC;         // 262144
  float* y2   = ws + off; off += (size_t)CB * CN * CFC;         // 262144
  float* Wz   = ws + off; off += (size_t)200 * 150;             // 30000
  (void)ws_size; (void)in_sizes; (void)n_in; (void)out_size;

  // One-time per call: dinv for all (b,t,n); pack Wz; init h = hidden0
  {
    int rows = CB * CT * CN;
    deg_dinv_kernel<<<(rows * 32 + 255) / 256, 256, 0, stream>>>(adj, dinv, rows, CN);
    pack_wz_kernel<<<(200 * 150 + 255) / 256, 256, 0, stream>>>(Wch, Wcu, Wz);
    hipMemcpyAsync(h, h0, sizeof(float) * CB * CN * CF, hipMemcpyDeviceToDevice, stream);
  }

  const int MN = CB * CN;  // 2048 flattened rows

  for (int t = 0; t < CT; ++t) {
    // Encoder: obs_t(512x10)@We1 + be1, ReLU  (batched over B: rows not contiguous)
    launch_gemm(obs + (long long)t * CN * CIN, (long long)CT * CN * CIN, CIN,
                We1, 0, CFC, tmp1, (long long)CN * CFC, CFC,
                be1, CN, CFC, CIN, CB, 1, stream);
    // tmp1(2048x128)@We2 + be2, ReLU -> x
    launch_gemm(tmp1, 0, CFC, We2, 0, CF, x, 0, CF,
                be2, MN, CF, CFC, 1, 1, stream);
    // Vs = dinv ⊙ [h, x]
    prescale_kernel<<<(MN * 2 * CF + 255) / 256, 256, 0, stream>>>(h, x, dinv, Vs, t);
    // W = A_t @ Vs   (the HBM-bound GEMM: 4 x (512x512)·(512x100))
    launch_gemm(adj + (long long)t * CN * CN, (long long)CT * CN * CN, CN,
                Vs, (long long)CN * 2 * CF, 2 * CF,
                Wb, (long long)CN * 2 * CF, 2 * CF,
                nullptr, CN, 2 * CF, CN, CB, 0, stream);
    // U = [h, Lh, x, Lx]
    combine_kernel<<<(MN * 4 * CF + 255) / 256, 256, 0, stream>>>(h, x, Wb, dinv, U, t);
    // z = U(2048x200) @ Wz(200x150) + bc
    launch_gemm(U, 0, 4 * CF, Wz, 0, 150, z, 0, 150,
                bc, MN, 150, 4 * CF, 1, 0, stream);
    // CfC gate update -> h
    gate_kernel<<<(MN * CF + 255) / 256, 256, 0, stream>>>(z, h);
    // Decoder
    launch_gemm(h, 0, CF, Wd1, 0, CFC, y1, 0, CFC, bd1, MN, CFC, CF, 1, 1, stream);
    launch_gemm(y1, 0, CFC, Wd2, 0, CFC, y2, 0, CFC, bd2, MN, CFC, CFC, 1, 1, stream);
    // Final layer writes straight into actions (B,T,N,2)
    launch_gemm(y2, (long long)CN * CFC, CFC, Wd3, 0, COUT,
                out + (long long)t * CN * COUT, (long long)CT * CN * COUT, COUT,
                bd3, CN, COUT, CFC, CB, 0, stream);
  }

  // h_final appended flat after actions
  hipMemcpyAsync(out + (long long)CB * CT * CN * COUT, h,
                 sizeof(float) * CB * CN * CF, hipMemcpyDeviceToDevice, stream);
}